// PrivateGNN_66975720014145
// MI455X (gfx1250) — compile-verified
//
#include <hip/hip_runtime.h>
#include <hip/hip_bf16.h>
#include <math.h>

typedef __attribute__((ext_vector_type(2))) float v2f;
typedef __attribute__((ext_vector_type(8))) float v8f;

#define KDIM 256
#define ROW_TILES 5                    // 5 x 16 = 80 rows per block (10000 = 125*80)
#define ROWS_PER_BLOCK (ROW_TILES * 16)
#define LDS_STRIDE 260                 // pad: stride mod 64 banks = 4 -> rows hit distinct banks

// ---------------------------------------------------------------------------
// GEMM: out[N x Mout] = diag(rowScale?) * A[N x 256] @ W[256 x Mout] + bias
// Block = 80 rows x 128 cols. Each wave owns one 16-col stripe and 5 row-tiles:
// B (global, L2-resident) is loaded once per k-step and reused by 5 independent
// V_WMMA_F32_16X16X4_F32 ops (disjoint accumulators -> XDL pipelining, B load
// traffic amortized 5x vs one-tile-per-wave).
// ---------------------------------------------------------------------------
__global__ __launch_bounds__(256)
void gemm_bias_wmma(const float* __restrict__ A, const float* __restrict__ W,
                    const float* __restrict__ bias, const float* __restrict__ rowScale,
                    float* __restrict__ out, int Mout) {
    __shared__ float As[ROWS_PER_BLOCK * LDS_STRIDE];

    const int tid     = threadIdx.x;
    const int rowBase = blockIdx.x * ROWS_PER_BLOCK;

    // Stage 80x256 A tile into LDS (thread = k, loop = row; coalesced),
    // fusing the mean-aggregation divide (per-row scale) when present.
    #pragma unroll 8
    for (int r = 0; r < ROWS_PER_BLOCK; ++r) {
        float v = A[(rowBase + r) * KDIM + tid];
        if (rowScale) v *= rowScale[rowBase + r];
        As[r * LDS_STRIDE + tid] = v;
    }
    __syncthreads();

    const int wave    = tid >> 5;              // 8 waves per block
    const int lane    = tid & 31;
    const int colBase = (blockIdx.y * 8 + wave) * 16;
    const int n       = lane & 15;             // B/C/D lane -> output column in tile
    const int m       = lane & 15;             // A lane -> row in tile
    const int khalf   = (lane >> 4) << 1;      // lanes 0-15: K+{0,1}; 16-31: K+{2,3}

    // C/D layout: VGPR i, lane l -> row = i + 8*(l>>4), col = l&15.
    const float bv = bias[colBase + n];
    v8f acc[ROW_TILES];
    #pragma unroll
    for (int t = 0; t < ROW_TILES; ++t)
        acc[t] = (v8f){bv, bv, bv, bv, bv, bv, bv, bv};

    const float* __restrict__ Wc = W + colBase + n;

    #pragma unroll 2
    for (int k0 = 0; k0 < KDIM; k0 += 4) {
        const int kk = k0 + khalf;
        v2f b;
        b.x = Wc[(kk)     * Mout];
        b.y = Wc[(kk + 1) * Mout];
        #pragma unroll
        for (int t = 0; t < ROW_TILES; ++t) {
            v2f a = *(const v2f*)&As[(t * 16 + m) * LDS_STRIDE + kk];  // ds_load_b64
            acc[t] = __builtin_amdgcn_wmma_f32_16x16x4_f32(
                         false, a, false, b, (short)0, acc[t], false, false);
        }
    }

    const int rowOff = (lane >> 4) * 8;
    #pragma unroll
    for (int t = 0; t < ROW_TILES; ++t) {
        #pragma unroll
        for (int i = 0; i < 8; ++i) {
            out[(rowBase + t * 16 + rowOff + i) * Mout + colBase + n] = acc[t][i];
        }
    }
}

// ---------------------------------------------------------------------------
// Edge-phase kernels (bandwidth-dominant part of the workload)
// ---------------------------------------------------------------------------
__global__ void fill_zero(float* __restrict__ p, int n) {
    int i = blockIdx.x * blockDim.x + threadIdx.x;
    if (i < n) p[i] = 0.0f;
}

__global__ void degree_count(const long long* __restrict__ dst, float* __restrict__ deg, int E) {
    int i = blockIdx.x * blockDim.x + threadIdx.x;
    if (i < E) atomicAdd(&deg[(int)dst[i]], 1.0f);
}

__global__ void degree_inv(float* __restrict__ deg, int N) {
    int i = blockIdx.x * blockDim.x + threadIdx.x;
    if (i < N) deg[i] = 1.0f / fmaxf(deg[i], 1.0f);
}

// 4 edges per 256-thread block; 64 lanes x float4 = 256 dims per edge.
// float4 gather (b128 loads), per-dword f32 atomic scatter (hardware minimum).
__global__ __launch_bounds__(256)
void aggregate(const float* __restrict__ h, const long long* __restrict__ src,
               const long long* __restrict__ dst, float* __restrict__ agg, int E) {
    const int e = blockIdx.x * 4 + (threadIdx.x >> 6);
    if (e >= E) return;
    const int d4 = (threadIdx.x & 63) * 4;
    const int s  = (int)src[e];
    const int d  = (int)dst[e];
    const float4 v = *(const float4*)&h[s * 256 + d4];
    float* o = &agg[d * 256 + d4];
    atomicAdd(o + 0, v.x);
    atomicAdd(o + 1, v.y);
    atomicAdd(o + 2, v.z);
    atomicAdd(o + 3, v.w);
}

// ---------------------------------------------------------------------------
// BatchNorm + SELU
// ---------------------------------------------------------------------------
__global__ __launch_bounds__(256)
void bn_stats(const float* __restrict__ h, float* __restrict__ sum,
              float* __restrict__ sumsq, int N) {
    const int c = threadIdx.x;     // column; reads coalesced across threads
    float s = 0.0f, q = 0.0f;
    for (int r = blockIdx.x; r < N; r += gridDim.x) {
        float v = h[r * 256 + c];
        s += v;
        q += v * v;
    }
    atomicAdd(&sum[c],   s);
    atomicAdd(&sumsq[c], q);
}

__global__ void bn_finalize(const float* __restrict__ sum, const float* __restrict__ sumsq,
                            const float* __restrict__ g, const float* __restrict__ be,
                            float* __restrict__ scale, float* __restrict__ shift, int N) {
    const int c      = threadIdx.x;
    const float invN = 1.0f / (float)N;
    const float mu   = sum[c] * invN;
    const float var  = sumsq[c] * invN - mu * mu;
    const float sc   = g[c] * rsqrtf(var + 1e-5f);
    scale[c] = sc;
    shift[c] = be[c] - mu * sc;
}

__device__ __forceinline__ float selu_f(float v) {
    const float lam = 1.0507009873554805f;
    const float alp = 1.6732632423543772f;
    return (v > 0.0f) ? lam * v : lam * alp * (expf(v) - 1.0f);
}

// float4-vectorized: 4 | 256, so a quad never crosses a row and scale/shift
// quads are contiguous.
__global__ __launch_bounds__(256)
void bn_selu(float* __restrict__ h, const float* __restrict__ scale,
             const float* __restrict__ shift, int totalQuads) {
    const int i = blockIdx.x * blockDim.x + threadIdx.x;
    if (i >= totalQuads) return;
    const int c = (i * 4) & 255;
    float4 v  = *(float4*)&h[i * 4];
    const float4 sc = *(const float4*)&scale[c];
    const float4 sh = *(const float4*)&shift[c];
    v.x = selu_f(v.x * sc.x + sh.x);
    v.y = selu_f(v.y * sc.y + sh.y);
    v.z = selu_f(v.z * sc.z + sh.z);
    v.w = selu_f(v.w * sc.w + sh.w);
    *(float4*)&h[i * 4] = v;
}

// ---------------------------------------------------------------------------
// Orchestration
// ---------------------------------------------------------------------------
extern "C" void kernel_launch(void* const* d_in, const int* in_sizes, int n_in,
                              void* d_out, int out_size, void* d_ws, size_t ws_size,
                              hipStream_t stream) {
    const float*     x   = (const float*)d_in[0];
    const long long* ei  = (const long long*)d_in[1];
    const float* W0  = (const float*)d_in[2];
    const float* b0  = (const float*)d_in[3];
    const float* W1  = (const float*)d_in[4];
    const float* b1  = (const float*)d_in[5];
    const float* W2  = (const float*)d_in[6];
    const float* b2  = (const float*)d_in[7];
    const float* W3  = (const float*)d_in[8];
    const float* b3  = (const float*)d_in[9];
    const float* g0  = (const float*)d_in[10];
    const float* be0 = (const float*)d_in[11];
    const float* g1  = (const float*)d_in[12];
    const float* be1 = (const float*)d_in[13];
    const float* g2  = (const float*)d_in[14];
    const float* be2 = (const float*)d_in[15];

    const int N = in_sizes[0] / 256;      // 10000 (= 125 * 80)
    const int E = in_sizes[1] / 2;        // 320000
    const long long* src = ei;
    const long long* dst = ei + E;

    // Workspace layout
    char* ws = (char*)d_ws;
    const size_t HBYTES = (size_t)N * 256 * sizeof(float);
    float* hA    = (float*)(ws);
    float* agg   = (float*)(ws + HBYTES);
    float* inv   = (float*)(ws + 2 * HBYTES);
    float* sums  = (float*)(ws + 2 * HBYTES + ((size_t)N * 4 + 255) / 256 * 256);
    float* sum   = sums;          // 256
    float* sumsq = sums + 256;    // 256
    float* scale = sums + 512;    // 256
    float* shift = sums + 768;    // 256

    const int NT = N * 256;
    const int NQ = NT / 4;
    const dim3 gemmGridH(N / ROWS_PER_BLOCK, 2);   // Mout = 256
    const dim3 gemmGridO(N / ROWS_PER_BLOCK, 1);   // Mout = 128
    const int aggBlocks = (E + 3) / 4;

    // Degrees (depend only on dst -> compute once, store reciprocal)
    fill_zero<<<(N + 255) / 256, 256, 0, stream>>>(inv, N);
    degree_count<<<(E + 255) / 256, 256, 0, stream>>>(dst, inv, E);
    degree_inv<<<(N + 255) / 256, 256, 0, stream>>>(inv, N);

    // ---- Layer 0: h = selu(bn(x @ W0 + b0)) ----
    gemm_bias_wmma<<<gemmGridH, 256, 0, stream>>>(x, W0, b0, nullptr, hA, 256);
    fill_zero<<<2, 256, 0, stream>>>(sum, 512);
    bn_stats<<<100, 256, 0, stream>>>(hA, sum, sumsq, N);
    bn_finalize<<<1, 256, 0, stream>>>(sum, sumsq, g0, be0, scale, shift, N);
    bn_selu<<<(NQ + 255) / 256, 256, 0, stream>>>(hA, scale, shift, NQ);

    // ---- MP layer 1 ----
    fill_zero<<<(NT + 255) / 256, 256, 0, stream>>>(agg, NT);
    aggregate<<<aggBlocks, 256, 0, stream>>>(hA, src, dst, agg, E);
    gemm_bias_wmma<<<gemmGridH, 256, 0, stream>>>(agg, W1, b1, inv, hA, 256);
    fill_zero<<<2, 256, 0, stream>>>(sum, 512);
    bn_stats<<<100, 256, 0, stream>>>(hA, sum, sumsq, N);
    bn_finalize<<<1, 256, 0, stream>>>(sum, sumsq, g1, be1, scale, shift, N);
    bn_selu<<<(NQ + 255) / 256, 256, 0, stream>>>(hA, scale, shift, NQ);

    // ---- MP layer 2 ----
    fill_zero<<<(NT + 255) / 256, 256, 0, stream>>>(agg, NT);
    aggregate<<<aggBlocks, 256, 0, stream>>>(hA, src, dst, agg, E);
    gemm_bias_wmma<<<gemmGridH, 256, 0, stream>>>(agg, W2, b2, inv, hA, 256);
    fill_zero<<<2, 256, 0, stream>>>(sum, 512);
    bn_stats<<<100, 256, 0, stream>>>(hA, sum, sumsq, N);
    bn_finalize<<<1, 256, 0, stream>>>(sum, sumsq, g2, be2, scale, shift, N);
    bn_selu<<<(NQ + 255) / 256, 256, 0, stream>>>(hA, scale, shift, NQ);

    // ---- Final dense: out = h @ W3 + b3 (D_OUT = 128) ----
    gemm_bias_wmma<<<gemmGridO, 256, 0, stream>>>(hA, W3, b3, nullptr, (float*)d_out, 128);
}